// get_shared_model_40321152975316
// MI455X (gfx1250) — compile-verified
//
#include <hip/hip_runtime.h>
#include <hip/hip_bf16.h>

typedef __attribute__((ext_vector_type(2))) float v2f;
typedef __attribute__((ext_vector_type(8))) float v8f;

#define B_ 8
#define N0_ 4096

// --------------------------------------------------------------------------
// transpose xyz [B,7,N] -> l0_pts [B,N,7] and l0_xyz [B,N,3]
// --------------------------------------------------------------------------
__global__ void k_transpose_pts(const float* __restrict__ xyz,
                                float* __restrict__ l0_xyz,
                                float* __restrict__ l0_pts) {
  int t = blockIdx.x * blockDim.x + threadIdx.x;
  int total = B_ * N0_ * 7;
  if (t >= total) return;
  int c = t % 7;
  int n = (t / 7) % N0_;
  int b = t / (7 * N0_);
  float v = xyz[((long)b * 7 + c) * N0_ + n];
  l0_pts[((long)b * N0_ + n) * 7 + c] = v;
  if (c < 3) l0_xyz[((long)b * N0_ + n) * 3 + c] = v;
}

// --------------------------------------------------------------------------
// farthest point sampling: one workgroup per batch, LDS argmax reduction
// --------------------------------------------------------------------------
__global__ void k_fps(const float* __restrict__ xyz, int N, int npoint,
                      int* __restrict__ idx_out, float* __restrict__ dist_ws) {
  const int T = 256;
  int b = blockIdx.x;
  int tid = threadIdx.x;
  __shared__ float sval[T];
  __shared__ int sidx[T];
  __shared__ int sfar;
  float* dist = dist_ws + (long)b * N;
  for (int i = tid; i < N; i += T) dist[i] = 1e10f;
  __syncthreads();
  int far = 0;
  for (int s = 0; s < npoint; ++s) {
    if (tid == 0) idx_out[b * npoint + s] = far;
    float cx = xyz[((long)b * N + far) * 3 + 0];
    float cy = xyz[((long)b * N + far) * 3 + 1];
    float cz = xyz[((long)b * N + far) * 3 + 2];
    float best = -1.0f; int besti = 0;
    for (int i = tid; i < N; i += T) {
      float dx = xyz[((long)b * N + i) * 3 + 0] - cx;
      float dy = xyz[((long)b * N + i) * 3 + 1] - cy;
      float dz = xyz[((long)b * N + i) * 3 + 2] - cz;
      float d = dx * dx + dy * dy + dz * dz;
      float dd = fminf(dist[i], d);
      dist[i] = dd;
      if (dd > best) { best = dd; besti = i; }
    }
    sval[tid] = best; sidx[tid] = besti;
    __syncthreads();
    for (int off = T / 2; off > 0; off >>= 1) {
      if (tid < off) {
        if (sval[tid + off] > sval[tid]) {
          sval[tid] = sval[tid + off];
          sidx[tid] = sidx[tid + off];
        }
      }
      __syncthreads();
    }
    if (tid == 0) sfar = sidx[0];
    __syncthreads();
    far = sfar;
    __syncthreads();
  }
}

// gather new_xyz[b,s,:] = xyz[b, fidx[b,s], :]
__global__ void k_gather_xyz(const float* __restrict__ xyz, const int* __restrict__ fidx,
                             float* __restrict__ newxyz, int N, int S) {
  int t = blockIdx.x * blockDim.x + threadIdx.x;
  int total = B_ * S * 3;
  if (t >= total) return;
  int c = t % 3;
  int s = (t / 3) % S;
  int b = t / (3 * S);
  int g = fidx[b * S + s];
  newxyz[((long)b * S + s) * 3 + c] = xyz[((long)b * N + g) * 3 + c];
}

// --------------------------------------------------------------------------
// ball query matching the reference semantics: first K in-range point
// indices in ascending order, padded with the first in-range index
// --------------------------------------------------------------------------
__global__ void k_query_ball(const float* __restrict__ xyz, const float* __restrict__ newxyz,
                             int N, int S, int K, float r2, int* __restrict__ gidx) {
  int t = blockIdx.x * blockDim.x + threadIdx.x;
  if (t >= B_ * S) return;
  int b = t / S;
  float cx = newxyz[(long)t * 3 + 0];
  float cy = newxyz[(long)t * 3 + 1];
  float cz = newxyz[(long)t * 3 + 2];
  int cnt = 0, first = 0; bool has = false;
  for (int j = 0; j < N; ++j) {
    if (cnt >= K) break;
    float dx = xyz[((long)b * N + j) * 3 + 0] - cx;
    float dy = xyz[((long)b * N + j) * 3 + 1] - cy;
    float dz = xyz[((long)b * N + j) * 3 + 2] - cz;
    float d = dx * dx + dy * dy + dz * dz;
    if (d <= r2) {
      if (!has) { first = j; has = true; }
      gidx[(long)t * K + cnt] = j;
      ++cnt;
    }
  }
  for (; cnt < K; ++cnt) gidx[(long)t * K + cnt] = first;
}

// build grouped GEMM input rows: [points[b,g,:C], xyz[b,g,:]-new_xyz[b,s,:]]
__global__ void k_group(const float* __restrict__ pts, const float* __restrict__ xyz,
                        const float* __restrict__ newxyz, const int* __restrict__ gidx,
                        int N, int S, int K, int C, float* __restrict__ out) {
  long t = (long)blockIdx.x * blockDim.x + threadIdx.x;
  int Cin = C + 3;
  long total = (long)B_ * S * K * Cin;
  if (t >= total) return;
  int c = (int)(t % Cin);
  long r = t / Cin;
  int k = (int)(r % K);
  long bs = r / K;
  int s = (int)(bs % S);
  int b = (int)(bs / S);
  int g = gidx[bs * K + k];
  float v;
  if (c < C) {
    v = pts[((long)b * N + g) * C + c];
  } else {
    int a = c - C;
    v = xyz[((long)b * N + g) * 3 + a] - newxyz[((long)b * S + s) * 3 + a];
  }
  out[r * Cin + c] = v;
}

// --------------------------------------------------------------------------
// WMMA f32 GEMM:  Y[M,N] = act(X[M,K] * W[N,K]^T + bias[N])
// One 16x16 output tile per wave32 via V_WMMA_F32_16X16X4_F32.
// A frag (16x4): lanes 0-15 hold M=lane, K={0,1}; lanes 16-31 K={2,3}.
// B frag (4x16): mirrored; D: row = vgpr + 8*(lane>=16), col = lane&15.
// Main K-loop is branch-free (address-clamped); only the final partial
// K-step uses post-load selects (v_cndmask), never exec-mask branches.
// --------------------------------------------------------------------------
__global__ void k_gemm_bias_act(const float* __restrict__ X, const float* __restrict__ W,
                                const float* __restrict__ bias, float* __restrict__ Y,
                                int M, int N, int K, int relu) {
  int wave = threadIdx.x >> 5;
  int lane = threadIdx.x & 31;
  int tiles_n = (N + 15) >> 4;
  int tiles_m = (M + 15) >> 4;
  int tile = blockIdx.x * 8 + wave;
  if (tile >= tiles_m * tiles_n) return;
  int tm = (tile / tiles_n) << 4;
  int tn = (tile % tiles_n) << 4;
  int l15 = lane & 15;
  int khalf = (lane >> 4) << 1;  // 0 or 2
  int arow = tm + l15;
  int bcol = tn + l15;
  // address clamps only: an OOB A-lane pollutes only its own output row,
  // an OOB B-lane only its own output column; both are masked at the store.
  if (arow >= M) arow = M - 1;
  int colok = (bcol < N);
  if (!colok) bcol = N - 1;
  const float* Xr = X + (long)arow * K + khalf;
  const float* Wr = W + (long)bcol * K + khalf;
  __builtin_prefetch(Xr, 0, 1);
  __builtin_prefetch(Wr, 0, 1);
  v8f acc = {0.f, 0.f, 0.f, 0.f, 0.f, 0.f, 0.f, 0.f};
  int Kmain = K & ~3;
#pragma unroll 2
  for (int k = 0; k < Kmain; k += 4) {
    float a0 = Xr[k];
    float a1 = Xr[k + 1];
    float b0 = Wr[k];
    float b1 = Wr[k + 1];
    v2f A = {a0, a1};
    v2f Bv = {b0, b1};
    acc = __builtin_amdgcn_wmma_f32_16x16x4_f32(false, A, false, Bv,
                                                (short)0, acc, false, false);
  }
  if (Kmain < K) {
    int k0 = Kmain + khalf;          // absolute K index of this lane's pair
    int kr0 = (k0 < K) ? Kmain : 0;  // clamped relative offsets (Xr/Wr
    int kr1 = (k0 + 1 < K) ? Kmain + 1 : 0;  //  already include khalf)
    float a0 = Xr[kr0];
    float a1 = Xr[kr1];
    float b0 = Wr[kr0];
    float b1 = Wr[kr1];
    if (k0 >= K)     { a0 = 0.f; b0 = 0.f; }
    if (k0 + 1 >= K) { a1 = 0.f; b1 = 0.f; }
    v2f A = {a0, a1};
    v2f Bv = {b0, b1};
    acc = __builtin_amdgcn_wmma_f32_16x16x4_f32(false, A, false, Bv,
                                                (short)0, acc, false, false);
  }
  int rbase = tm + ((lane >> 4) << 3);
  int col = tn + l15;
  float bv = bias[col < N ? col : N - 1];
#pragma unroll
  for (int i = 0; i < 8; ++i) {
    int r = rbase + i;
    if (r < M && col < N) {
      float v = acc[i] + bv;
      if (relu) v = fmaxf(v, 0.f);
      Y[(long)r * N + col] = v;
    }
  }
}

// max over the K samples of a group: dst[b,s,dstOff+c] = max_k in[(b,s,k),c]
__global__ void k_maxpool(const float* __restrict__ in, float* __restrict__ dst,
                          int S, int K, int C, int dstC, int dstOff) {
  int t = blockIdx.x * blockDim.x + threadIdx.x;
  int total = B_ * S * C;
  if (t >= total) return;
  int c = t % C;
  int s = (t / C) % S;
  int b = t / (C * S);
  const float* p = in + ((long)(b * S + s) * K) * C + c;
  float m = p[0];
  for (int k = 1; k < K; ++k) m = fmaxf(m, p[(long)k * C]);
  dst[((long)b * S + s) * dstC + dstOff + c] = m;
}

// --------------------------------------------------------------------------
// feature propagation: 3-NN inverse-squared-distance interpolation + concat
// --------------------------------------------------------------------------
__global__ void k_fp_interp(const float* __restrict__ xyz1, const float* __restrict__ xyz2,
                            const float* __restrict__ p1, const float* __restrict__ p2,
                            int Np, int Mp, int C1, int C2, float* __restrict__ out) {
  int t = blockIdx.x * blockDim.x + threadIdx.x;
  if (t >= B_ * Np) return;
  int n = t % Np;
  int b = t / Np;
  float qx = xyz1[((long)b * Np + n) * 3 + 0];
  float qy = xyz1[((long)b * Np + n) * 3 + 1];
  float qz = xyz1[((long)b * Np + n) * 3 + 2];
  float d0 = 1e30f, d1 = 1e30f, d2 = 1e30f;
  int i0 = 0, i1 = 0, i2 = 0;
  for (int j = 0; j < Mp; ++j) {
    float dx = xyz2[((long)b * Mp + j) * 3 + 0] - qx;
    float dy = xyz2[((long)b * Mp + j) * 3 + 1] - qy;
    float dz = xyz2[((long)b * Mp + j) * 3 + 2] - qz;
    float d = dx * dx + dy * dy + dz * dz;
    if (d < d0)      { d2 = d1; i2 = i1; d1 = d0; i1 = i0; d0 = d; i0 = j; }
    else if (d < d1) { d2 = d1; i2 = i1; d1 = d;  i1 = j; }
    else if (d < d2) { d2 = d;  i2 = j; }
  }
  float w0 = 1.f / (d0 + 1e-8f);
  float w1 = 1.f / (d1 + 1e-8f);
  float w2 = 1.f / (d2 + 1e-8f);
  float ws = w0 + w1 + w2;
  w0 /= ws; w1 /= ws; w2 /= ws;
  float* o = out + (long)t * (C1 + C2);
  for (int c = 0; c < C1; ++c) o[c] = p1[((long)b * Np + n) * C1 + c];
  const float* q0 = p2 + ((long)b * Mp + i0) * C2;
  const float* q1 = p2 + ((long)b * Mp + i1) * C2;
  const float* q2 = p2 + ((long)b * Mp + i2) * C2;
  for (int c = 0; c < C2; ++c)
    o[C1 + c] = w0 * q0[c] + w1 * q1[c] + w2 * q2[c];
}

// --------------------------------------------------------------------------
// fused heads: contact(2), normal(3, L2-normalized), force(1) from x[.,128]
// --------------------------------------------------------------------------
__global__ void k_head(const float* __restrict__ x,
                       const float* __restrict__ Wc, const float* __restrict__ bc,
                       const float* __restrict__ Wn, const float* __restrict__ bn,
                       const float* __restrict__ Wf, const float* __restrict__ bf,
                       float* __restrict__ contact, float* __restrict__ normal,
                       float* __restrict__ force) {
  int t = blockIdx.x * blockDim.x + threadIdx.x;
  if (t >= B_ * N0_) return;
  const float* xr = x + (long)t * 128;
  float c0 = bc[0], c1 = bc[1];
  float n0 = bn[0], n1 = bn[1], n2 = bn[2];
  float f0 = bf[0];
  for (int k = 0; k < 128; ++k) {
    float xv = xr[k];
    c0 += xv * Wc[k];
    c1 += xv * Wc[128 + k];
    n0 += xv * Wn[k];
    n1 += xv * Wn[128 + k];
    n2 += xv * Wn[256 + k];
    f0 += xv * Wf[k];
  }
  float nrm = fmaxf(sqrtf(n0 * n0 + n1 * n1 + n2 * n2), 1e-12f);
  contact[(long)t * 2 + 0] = c0;
  contact[(long)t * 2 + 1] = c1;
  normal[(long)t * 3 + 0] = n0 / nrm;
  normal[(long)t * 3 + 1] = n1 / nrm;
  normal[(long)t * 3 + 2] = n2 / nrm;
  force[t] = f0;
}

// l4_points = transpose(l4p [B,16,1024], (0,2,1)) -> [B,1024,16]
__global__ void k_l4t(const float* __restrict__ l4p, float* __restrict__ out) {
  int t = blockIdx.x * blockDim.x + threadIdx.x;
  int total = B_ * 16 * 1024;
  if (t >= total) return;
  int c = t % 1024;
  int s = (t / 1024) % 16;
  int b = t / (1024 * 16);
  out[((long)b * 1024 + c) * 16 + s] = l4p[t];
}

// --------------------------------------------------------------------------
// host orchestration
// --------------------------------------------------------------------------
extern "C" void kernel_launch(void* const* d_in, const int* in_sizes, int n_in,
                              void* d_out, int out_size, void* d_ws, size_t ws_size,
                              hipStream_t stream) {
  (void)in_sizes; (void)n_in; (void)out_size; (void)ws_size;
  const float* xyz = (const float*)d_in[0];

  // workspace carving (256B aligned)
  char* ws = (char*)d_ws;
  size_t off = 0;
  auto allocF = [&](size_t n) -> float* {
    float* p = (float*)(ws + off);
    off = (off + n * sizeof(float) + 255) & ~(size_t)255;
    return p;
  };
  auto allocI = [&](size_t n) -> int* {
    int* p = (int*)(ws + off);
    off = (off + n * sizeof(int) + 255) & ~(size_t)255;
    return p;
  };

  float* l0_xyz = allocF((size_t)B_ * N0_ * 3);
  float* l0_pts = allocF((size_t)B_ * N0_ * 7);
  float* l1_xyz = allocF((size_t)B_ * 1024 * 3);
  float* l1_pts = allocF((size_t)B_ * 1024 * 96);
  float* l2_xyz = allocF((size_t)B_ * 256 * 3);
  float* l2_pts = allocF((size_t)B_ * 256 * 256);
  float* l3_xyz = allocF((size_t)B_ * 64 * 3);
  float* l3_pts = allocF((size_t)B_ * 64 * 512);
  float* l4_xyz = allocF((size_t)B_ * 16 * 3);
  float* l4_pts = allocF((size_t)B_ * 16 * 1024);
  float* l3_new = allocF((size_t)B_ * 64 * 256);
  float* l2_new = allocF((size_t)B_ * 256 * 256);
  float* l1_new = allocF((size_t)B_ * 1024 * 128);
  float* l0_new = allocF((size_t)B_ * N0_ * 128);
  float* xhead  = allocF((size_t)B_ * N0_ * 128);
  float* fdist  = allocF((size_t)B_ * N0_);
  int*   fidx   = allocI((size_t)B_ * 1024);
  int*   gidx   = allocI((size_t)B_ * 1024 * 32);
  float* bufA   = allocF((size_t)262144 * 64);
  float* bufB   = allocF((size_t)262144 * 64);

  auto gemm = [&](const float* X, int widx, int M, int N, int K, int relu, float* Y) {
    int tiles = ((M + 15) / 16) * ((N + 15) / 16);
    int blocks = (tiles + 7) / 8;
    k_gemm_bias_act<<<blocks, 256, 0, stream>>>(
        X, (const float*)d_in[widx], (const float*)d_in[widx + 1], Y, M, N, K, relu);
  };

  // one SA branch: ball query, group, MLP (WMMA), maxpool into lout channels
  auto sa_branch = [&](const float* xin, const float* pin, int N, int C,
                       const float* newxyz, int S, float radius, int K,
                       int widx0, const int* widths, int nlayers,
                       float* lout, int loutC, int loutOff) {
    int bs = B_ * S;
    k_query_ball<<<(bs + 255) / 256, 256, 0, stream>>>(xin, newxyz, N, S, K,
                                                       radius * radius, gidx);
    long rows = (long)bs * K;
    int Cin = C + 3;
    long total = rows * Cin;
    k_group<<<(int)((total + 255) / 256), 256, 0, stream>>>(pin, xin, newxyz, gidx,
                                                            N, S, K, C, bufA);
    float* src = bufA;
    int k = Cin;
    for (int l = 0; l < nlayers; ++l) {
      float* dst = (src == bufA) ? bufB : bufA;
      gemm(src, widx0 + 2 * l, (int)rows, widths[l], k, 1, dst);
      k = widths[l];
      src = dst;
    }
    int tp = B_ * S * k;
    k_maxpool<<<(tp + 255) / 256, 256, 0, stream>>>(src, lout, S, K, k, loutC, loutOff);
  };

  // one FP module: 3-NN interp + concat, then MLP; last layer -> outFinal
  auto fp = [&](const float* x1, const float* x2, const float* p1, const float* p2,
                int Np, int Mp, int C1, int C2, int widx0, const int* widths,
                int nlayers, float* outFinal) {
    int rows = B_ * Np;
    k_fp_interp<<<(rows + 255) / 256, 256, 0, stream>>>(x1, x2, p1, p2, Np, Mp,
                                                        C1, C2, bufA);
    float* src = bufA;
    int k = C1 + C2;
    for (int l = 0; l < nlayers; ++l) {
      float* dst = (l == nlayers - 1) ? outFinal : ((src == bufA) ? bufB : bufA);
      gemm(src, widx0 + 2 * l, rows, widths[l], k, 1, dst);
      k = widths[l];
      src = dst;
    }
  };

  // l0
  {
    int total = B_ * N0_ * 7;
    k_transpose_pts<<<(total + 255) / 256, 256, 0, stream>>>(xyz, l0_xyz, l0_pts);
  }

  static const int w_sa1b0[3] = {16, 16, 32};
  static const int w_sa1b1[3] = {32, 32, 64};
  static const int w_sa2b0[3] = {64, 64, 128};
  static const int w_sa2b1[3] = {64, 96, 128};
  static const int w_sa3b[3]  = {128, 196, 256};
  static const int w_sa4b0[3] = {256, 256, 512};
  static const int w_sa4b1[3] = {256, 384, 512};
  static const int w_fp44[2]  = {256, 256};
  static const int w_fp3[2]   = {256, 256};
  static const int w_fp2[2]   = {256, 128};
  static const int w_fp1[3]   = {128, 128, 128};

  // SA1: 4096 -> 1024
  k_fps<<<B_, 256, 0, stream>>>(l0_xyz, N0_, 1024, fidx, fdist);
  k_gather_xyz<<<(B_ * 1024 * 3 + 255) / 256, 256, 0, stream>>>(l0_xyz, fidx, l1_xyz, N0_, 1024);
  sa_branch(l0_xyz, l0_pts, N0_, 7, l1_xyz, 1024, 0.05f, 16, 1, w_sa1b0, 3, l1_pts, 96, 0);
  sa_branch(l0_xyz, l0_pts, N0_, 7, l1_xyz, 1024, 0.10f, 32, 7, w_sa1b1, 3, l1_pts, 96, 32);

  // SA2: 1024 -> 256
  k_fps<<<B_, 256, 0, stream>>>(l1_xyz, 1024, 256, fidx, fdist);
  k_gather_xyz<<<(B_ * 256 * 3 + 255) / 256, 256, 0, stream>>>(l1_xyz, fidx, l2_xyz, 1024, 256);
  sa_branch(l1_xyz, l1_pts, 1024, 96, l2_xyz, 256, 0.10f, 16, 13, w_sa2b0, 3, l2_pts, 256, 0);
  sa_branch(l1_xyz, l1_pts, 1024, 96, l2_xyz, 256, 0.20f, 32, 19, w_sa2b1, 3, l2_pts, 256, 128);

  // SA3: 256 -> 64
  k_fps<<<B_, 256, 0, stream>>>(l2_xyz, 256, 64, fidx, fdist);
  k_gather_xyz<<<(B_ * 64 * 3 + 255) / 256, 256, 0, stream>>>(l2_xyz, fidx, l3_xyz, 256, 64);
  sa_branch(l2_xyz, l2_pts, 256, 256, l3_xyz, 64, 0.20f, 16, 25, w_sa3b, 3, l3_pts, 512, 0);
  sa_branch(l2_xyz, l2_pts, 256, 256, l3_xyz, 64, 0.40f, 32, 31, w_sa3b, 3, l3_pts, 512, 256);

  // SA4: 64 -> 16
  k_fps<<<B_, 256, 0, stream>>>(l3_xyz, 64, 16, fidx, fdist);
  k_gather_xyz<<<(B_ * 16 * 3 + 255) / 256, 256, 0, stream>>>(l3_xyz, fidx, l4_xyz, 64, 16);
  sa_branch(l3_xyz, l3_pts, 64, 512, l4_xyz, 16, 0.40f, 16, 37, w_sa4b0, 3, l4_pts, 1024, 0);
  sa_branch(l3_xyz, l3_pts, 64, 512, l4_xyz, 16, 0.80f, 32, 43, w_sa4b1, 3, l4_pts, 1024, 512);

  // FP modules
  fp(l3_xyz, l4_xyz, l3_pts, l4_pts, 64, 16, 512, 1024, 49, w_fp44, 2, l3_new);
  fp(l2_xyz, l3_xyz, l2_pts, l3_new, 256, 64, 256, 256, 53, w_fp3, 2, l2_new);
  fp(l1_xyz, l2_xyz, l1_pts, l2_new, 1024, 256, 96, 256, 57, w_fp2, 2, l1_new);
  fp(l0_xyz, l1_xyz, nullptr, l1_new, N0_, 1024, 0, 128, 61, w_fp1, 3, l0_new);

  // head
  gemm(l0_new, 67, B_ * N0_, 128, 128, 1, xhead);
  float* out = (float*)d_out;
  float* out_contact = out;
  float* out_normal = out + (size_t)B_ * N0_ * 2;
  float* out_force = out_normal + (size_t)B_ * N0_ * 3;
  float* out_l4 = out_force + (size_t)B_ * N0_ * 1;
  k_head<<<(B_ * N0_ + 255) / 256, 256, 0, stream>>>(
      xhead,
      (const float*)d_in[69], (const float*)d_in[70],
      (const float*)d_in[71], (const float*)d_in[72],
      (const float*)d_in[73], (const float*)d_in[74],
      out_contact, out_normal, out_force);
  k_l4t<<<(B_ * 16 * 1024 + 255) / 256, 256, 0, stream>>>(l4_pts, out_l4);
}